// Resample2d_11304353923109
// MI455X (gfx1250) — compile-verified
//
#include <hip/hip_runtime.h>

// Problem geometry (fixed by the reference):
//   input1: (8, 8, 512, 1024) f32   image
//   input2: (8, 2, 512, 1024) f32   flow (fx plane, fy plane)
//   output: (8, 8, 512, 1024) f32
#define RS_B 8
#define RS_C 8
#define RS_H 512
#define RS_W 1024
#define RS_PLANE (RS_H * RS_W)   // 524288 elements per (b,c) plane

__global__ __launch_bounds__(256) void resample2d_bilinear_kernel(
    const float* __restrict__ img,   // input1
    const float* __restrict__ flow,  // input2
    float* __restrict__ out)
{
    // One thread per (b, y, x); channels handled in-thread (index/weight reuse).
    const int p = blockIdx.x * 256 + threadIdx.x;   // pixel index within a plane
    const int b = blockIdx.y;
    const int x = p & (RS_W - 1);
    const int y = p >> 10;                          // W == 1024

    // Flow is read exactly once across the whole kernel -> non-temporal loads
    // (th:TH_LOAD_NT) so it does not evict the reused image from L2.
    const float fx = __builtin_nontemporal_load(flow + (size_t)(b * 2 + 0) * RS_PLANE + p);
    const float fy = __builtin_nontemporal_load(flow + (size_t)(b * 2 + 1) * RS_PLANE + p);

    const float x2 = (float)x + fx;
    const float y2 = (float)y + fy;

    // astype(int32) truncates toward zero; C++ (int) cast matches exactly.
    int ix_l = (int)x2;
    int iy_t = (int)y2;
    ix_l = ix_l < 0 ? 0 : (ix_l > RS_W - 1 ? RS_W - 1 : ix_l);
    iy_t = iy_t < 0 ? 0 : (iy_t > RS_H - 1 ? RS_H - 1 : iy_t);
    const int ix_r = (ix_l + 1 > RS_W - 1) ? RS_W - 1 : ix_l + 1;
    const int iy_b = (iy_t + 1 > RS_H - 1) ? RS_H - 1 : iy_t + 1;

    // Weights from the *clipped* integer corner (matches reference; alpha/beta
    // may be negative at the low edge).
    const float a  = x2 - (float)ix_l;
    const float bb = y2 - (float)iy_t;
    const float w_tl = (1.0f - a) * (1.0f - bb);
    const float w_tr = a * (1.0f - bb);
    const float w_bl = (1.0f - a) * bb;
    const float w_br = a * bb;

    const int o_tl = iy_t * RS_W + ix_l;
    const int o_tr = iy_t * RS_W + ix_r;
    const int o_bl = iy_b * RS_W + ix_l;
    const int o_br = iy_b * RS_W + ix_r;

    const size_t base = (size_t)b * RS_C * RS_PLANE;

    // Fully unrolled: 32 temporal gather loads in flight (deep MLP), then
    // 8 non-temporal output stores (write-once data bypasses L2 retention).
#pragma unroll
    for (int c = 0; c < RS_C; ++c) {
        const float* pl = img + base + (size_t)c * RS_PLANE;
        const float tl = pl[o_tl];
        const float tr = pl[o_tr];
        const float bl = pl[o_bl];
        const float br = pl[o_br];
        float v = w_tl * tl;
        v = fmaf(w_tr, tr, v);
        v = fmaf(w_bl, bl, v);
        v = fmaf(w_br, br, v);
        __builtin_nontemporal_store(v, out + base + (size_t)c * RS_PLANE + p);
    }
}

extern "C" void kernel_launch(void* const* d_in, const int* in_sizes, int n_in,
                              void* d_out, int out_size, void* d_ws, size_t ws_size,
                              hipStream_t stream) {
    (void)in_sizes; (void)n_in; (void)out_size; (void)d_ws; (void)ws_size;
    const float* img  = (const float*)d_in[0];   // input1: 8*8*512*1024 f32
    const float* flow = (const float*)d_in[1];   // input2: 8*2*512*1024 f32
    float* out = (float*)d_out;

    dim3 block(256, 1, 1);
    dim3 grid(RS_PLANE / 256, RS_B, 1);          // (2048, 8) -> 4M threads
    resample2d_bilinear_kernel<<<grid, block, 0, stream>>>(img, flow, out);
}